// VulnerabilityGNN_47476568490190
// MI455X (gfx1250) — compile-verified
//
#include <hip/hip_runtime.h>
#include <hip/hip_bf16.h>

typedef __attribute__((ext_vector_type(2))) float v2f;
typedef __attribute__((ext_vector_type(8))) float v8f;

#define HID     64
#define HEADS   4
#define FIN     128
#define NGRAPH  512
#define CLASSES 2

// order-preserving float<->uint encoding for atomicMax on floats
__device__ __forceinline__ unsigned fenc(float f) {
    unsigned u = __float_as_uint(f);
    return (u & 0x80000000u) ? ~u : (u | 0x80000000u);
}
__device__ __forceinline__ float fdec(unsigned u) {
    return __uint_as_float((u & 0x80000000u) ? (u & 0x7FFFFFFFu) : ~u);
}
#define ENC_NEG_INF 0x007FFFFFu  // fenc(-inf)

// ---------------- init helpers ----------------
__global__ void k_fset(float* p, float v, int n) {
    int i = blockIdx.x * blockDim.x + threadIdx.x;
    if (i < n) p[i] = v;
}
__global__ void k_uset(unsigned* p, unsigned v, int n) {
    int i = blockIdx.x * blockDim.x + threadIdx.x;
    if (i < n) p[i] = v;
}

// ---------------- degree / norm ----------------
__global__ void k_count_deg(const int* __restrict__ ei, float* deg, int E) {
    int t = blockIdx.x * blockDim.x + threadIdx.x;
    if (t < E) atomicAdd(&deg[ei[E + t]], 1.0f);
}
__global__ void k_dinv(float* d, int N) {
    int i = blockIdx.x * blockDim.x + threadIdx.x;
    if (i < N) d[i] = rsqrtf(fmaxf(d[i], 1.0f));
}

// ---------------- fp32 WMMA GEMM: OUT[N x C] = X[N x K] @ W[K x C] ------------
// One wave per 16x16 output tile; K stepped by 4 via V_WMMA_F32_16X16X4_F32.
__global__ __launch_bounds__(256) void k_gemm_f32_wmma(
    const float* __restrict__ X, const float* __restrict__ W,
    float* __restrict__ OUT, int Nrows, int K, int C) {
    int lane   = threadIdx.x & 31;
    int wave   = threadIdx.x >> 5;
    int tilesN = C >> 4;
    int tilesM = (Nrows + 15) >> 4;
    int tile   = blockIdx.x * 8 + wave;
    if (tile >= tilesM * tilesN) return;   // wave-uniform exit: EXEC stays all-1s
    int tM = tile / tilesN, tN = tile % tilesN;
    int m0 = tM << 4, n0 = tN << 4;
    int r    = lane & 15;          // M (for A) / N (for B,C,D) within tile
    int half = lane >> 4;          // lane group 0/1
    int koff = half << 1;          // K offset 0 or 2 for A/B fragments

    int row = m0 + r;
    if (row >= Nrows) row = Nrows - 1;     // clamp loads; stores are guarded
    const float* xrow = X + (long)row * K;
    int col = n0 + r;

    v8f acc = {0.f, 0.f, 0.f, 0.f, 0.f, 0.f, 0.f, 0.f};
    for (int k = 0; k < K; k += 4) {
        v2f a = *(const v2f*)(xrow + k + koff);            // A[M][k+koff, k+koff+1]
        v2f b;
        b.x = W[(long)(k + koff)     * C + col];           // B[k+koff  ][N]
        b.y = W[(long)(k + koff + 1) * C + col];           // B[k+koff+1][N]
        acc = __builtin_amdgcn_wmma_f32_16x16x4_f32(
            /*neg_a=*/false, a, /*neg_b=*/false, b,
            /*c_mod=*/(short)0, acc, /*reuse_a=*/false, /*reuse_b=*/false);
    }
#pragma unroll
    for (int i = 0; i < 8; i++) {
        int rr = m0 + i + (half << 3);     // D vgpr i holds M = i (+8 for lanes 16-31)
        if (rr < Nrows) OUT[(long)rr * C + col] = acc[i];
    }
}

// ---------------- GCN aggregation ----------------
// self-loop contribution also initializes the accumulator
__global__ void k_gcn_self(const float* __restrict__ h, const float* __restrict__ dinv,
                           float* __restrict__ xout, int N) {
    int i = blockIdx.x * blockDim.x + threadIdx.x;
    if (i < N * HID) {
        int n = i >> 6;
        float s = dinv[n];
        xout[i] = h[i] * s * s;
    }
}
// one wave per edge: 64 features, 2 per lane
__global__ __launch_bounds__(256) void k_gcn_edges(
    const int* __restrict__ ei, const float* __restrict__ h,
    const float* __restrict__ dinv, float* __restrict__ xout, int E) {
    int lane = threadIdx.x & 31;
    int e = (blockIdx.x * blockDim.x + threadIdx.x) >> 5;
    if (e >= E) return;
    int s = ei[e], d = ei[E + e];
    float norm = dinv[s] * dinv[d];
    const float* hs = h + (long)s * HID;
    float* xd = xout + (long)d * HID;
    atomicAdd(&xd[lane],      hs[lane]      * norm);
    atomicAdd(&xd[lane + 32], hs[lane + 32] * norm);
}
__global__ void k_relu_bias(float* x, const float* __restrict__ b, int total) {
    int i = blockIdx.x * blockDim.x + threadIdx.x;
    if (i < total) x[i] = fmaxf(x[i] + b[i & (HID - 1)], 0.0f);
}

// ---------------- GAT ----------------
__global__ void k_gat_scores(const float* __restrict__ h,
                             const float* __restrict__ att_s, const float* __restrict__ att_d,
                             float* __restrict__ asrc, float* __restrict__ adst, int N) {
    int i = blockIdx.x * blockDim.x + threadIdx.x;
    if (i >= N * HEADS) return;
    int n = i >> 2, hh = i & 3;
    const float* hp = h + (long)n * (HEADS * HID) + hh * HID;
    const float* ap = att_s + hh * HID;
    const float* bp = att_d + hh * HID;
    float s0 = 0.f, s1 = 0.f;
    for (int c = 0; c < HID; c++) { float v = hp[c]; s0 += v * ap[c]; s1 += v * bp[c]; }
    asrc[i] = s0; adst[i] = s1;
}
__device__ __forceinline__ void edge_sd(const int* ei, int t, int E, int& s, int& d) {
    if (t < E) { s = ei[t]; d = ei[E + t]; } else { s = d = t - E; }  // self loops
}
__global__ void k_gat_max(const int* __restrict__ ei, const float* __restrict__ asrc,
                          const float* __restrict__ adst, unsigned* __restrict__ menc,
                          int E, int N) {
    int i = blockIdx.x * blockDim.x + threadIdx.x;
    if (i >= (E + N) * HEADS) return;
    int t = i >> 2, hh = i & 3, s, d;
    edge_sd(ei, t, E, s, d);
    float e = asrc[s * 4 + hh] + adst[d * 4 + hh];
    e = e > 0.f ? e : 0.2f * e;
    atomicMax(&menc[d * 4 + hh], fenc(e));
}
__global__ void k_gat_denom(const int* __restrict__ ei, const float* __restrict__ asrc,
                            const float* __restrict__ adst, const unsigned* __restrict__ menc,
                            float* __restrict__ denom, int E, int N) {
    int i = blockIdx.x * blockDim.x + threadIdx.x;
    if (i >= (E + N) * HEADS) return;
    int t = i >> 2, hh = i & 3, s, d;
    edge_sd(ei, t, E, s, d);
    float e = asrc[s * 4 + hh] + adst[d * 4 + hh];
    e = e > 0.f ? e : 0.2f * e;
    atomicAdd(&denom[d * 4 + hh], expf(e - fdec(menc[d * 4 + hh])));
}
// weighted aggregation with head-mean folded in: one wave per (edge|self)
__global__ __launch_bounds__(256) void k_gat_aggr(
    const int* __restrict__ ei, const float* __restrict__ h,
    const float* __restrict__ asrc, const float* __restrict__ adst,
    const unsigned* __restrict__ menc, const float* __restrict__ denom,
    float* __restrict__ xout, int E, int N) {
    int lane = threadIdx.x & 31;
    int t = (blockIdx.x * blockDim.x + threadIdx.x) >> 5;
    if (t >= E + N) return;
    int s, d;
    edge_sd(ei, t, E, s, d);
    float alpha[HEADS];
#pragma unroll
    for (int hh = 0; hh < HEADS; hh++) {
        float e = asrc[s * 4 + hh] + adst[d * 4 + hh];
        e = e > 0.f ? e : 0.2f * e;
        alpha[hh] = expf(e - fdec(menc[d * 4 + hh])) / denom[d * 4 + hh];
    }
    const float* hp = h + (long)s * (HEADS * HID);
    float* xd = xout + (long)d * HID;
#pragma unroll
    for (int c = lane; c < HID; c += 32) {
        float acc = 0.f;
#pragma unroll
        for (int hh = 0; hh < HEADS; hh++) acc += alpha[hh] * hp[hh * HID + c];
        atomicAdd(&xd[c], 0.25f * acc);
    }
}

// ---------------- pooling ----------------
__global__ void k_pool_cnt(const int* __restrict__ batch, float* cnt, int N) {
    int n = blockIdx.x * blockDim.x + threadIdx.x;
    if (n < N) atomicAdd(&cnt[batch[n]], 1.0f);
}
__global__ void k_pool(const int* __restrict__ batch, const float* __restrict__ x,
                       float* __restrict__ pmean, unsigned* __restrict__ pmax, int N) {
    int i = blockIdx.x * blockDim.x + threadIdx.x;
    if (i >= N * HID) return;
    int n = i >> 6, c = i & 63;
    int b = batch[n];
    float v = x[i];
    atomicAdd(&pmean[b * HID + c], v);
    atomicMax(&pmax[b * HID + c], fenc(v));
}
__global__ void k_pool_finish(const float* __restrict__ pmean, const unsigned* __restrict__ pmax,
                              const float* __restrict__ pcnt, float* __restrict__ g, int B) {
    int i = blockIdx.x * blockDim.x + threadIdx.x;
    if (i >= B * HID) return;
    int b = i >> 6;
    float cnt = pcnt[b];
    float mean = pmean[i] / fmaxf(cnt, 1.0f);
    float mx = (cnt > 0.f) ? fdec(pmax[i]) : 0.f;
    g[i] = mean + mx;
}

// ---------------- MLP head ----------------
__global__ void k_mlp(const float* __restrict__ g,
                      const float* __restrict__ w1, const float* __restrict__ b1,
                      const float* __restrict__ w2, const float* __restrict__ b2,
                      float* __restrict__ out, int B) {
    int b = blockIdx.x * blockDim.x + threadIdx.x;
    if (b >= B) return;
    const float* gr = g + b * HID;
    float hid[HID / 2];
#pragma unroll 4
    for (int j = 0; j < HID / 2; j++) {
        float acc = b1[j];
        for (int c = 0; c < HID; c++) acc += gr[c] * w1[c * (HID / 2) + j];
        hid[j] = fmaxf(acc, 0.f);
    }
    for (int j = 0; j < CLASSES; j++) {
        float acc = b2[j];
        for (int c = 0; c < HID / 2; c++) acc += hid[c] * w2[c * CLASSES + j];
        out[b * CLASSES + j] = acc;
    }
}

// ---------------- launch ----------------
extern "C" void kernel_launch(void* const* d_in, const int* in_sizes, int n_in,
                              void* d_out, int out_size, void* d_ws, size_t ws_size,
                              hipStream_t stream) {
    const float* x_in   = (const float*)d_in[0];
    const int*   ei     = (const int*)d_in[1];
    const int*   batch  = (const int*)d_in[2];
    const float* w0     = (const float*)d_in[3];
    const float* b0     = (const float*)d_in[4];
    const float* w1     = (const float*)d_in[5];
    const float* b1     = (const float*)d_in[6];
    const float* w2     = (const float*)d_in[7];
    const float* b2     = (const float*)d_in[8];
    const float* gat_w  = (const float*)d_in[9];
    const float* att_s  = (const float*)d_in[10];
    const float* att_d  = (const float*)d_in[11];
    const float* gat_b  = (const float*)d_in[12];
    const float* l1w    = (const float*)d_in[13];
    const float* l1b    = (const float*)d_in[14];
    const float* l2w    = (const float*)d_in[15];
    const float* l2b    = (const float*)d_in[16];
    float* out = (float*)d_out;

    const int N = in_sizes[2];        // batch vector length == #nodes
    const int E = in_sizes[1] / 2;

    // workspace layout (floats)
    float* ws = (float*)d_ws;
    size_t o = 0;
    float*    dinv  = ws + o; o += (size_t)N;
    float*    hbuf  = ws + o; o += (size_t)N * HEADS * HID;   // up to N x 256
    float*    xa    = ws + o; o += (size_t)N * HID;
    float*    xb    = ws + o; o += (size_t)N * HID;
    float*    asrc  = ws + o; o += (size_t)N * HEADS;
    float*    adst  = ws + o; o += (size_t)N * HEADS;
    unsigned* menc  = (unsigned*)(ws + o); o += (size_t)N * HEADS;
    float*    denom = ws + o; o += (size_t)N * HEADS;
    float*    pmean = ws + o; o += (size_t)NGRAPH * HID;
    unsigned* pmax  = (unsigned*)(ws + o); o += (size_t)NGRAPH * HID;
    float*    pcnt  = ws + o; o += (size_t)NGRAPH;
    float*    gbuf  = ws + o; o += (size_t)NGRAPH * HID;
    (void)ws_size; (void)n_in; (void)out_size;

    const int T = 256;
    auto blks = [](long n, int t) { return (int)((n + t - 1) / t); };

    // degree -> dinv
    k_fset<<<blks(N, T), T, 0, stream>>>(dinv, 1.0f, N);               // self-loop
    k_count_deg<<<blks(E, T), T, 0, stream>>>(ei, dinv, E);
    k_dinv<<<blks(N, T), T, 0, stream>>>(dinv, N);

    // 3 GCN layers (ping-pong xa/xb)
    const float* lx[3] = { x_in, xa, xb };
    float*       ly[3] = { xa, xb, xa };
    const float* lw[3] = { w0, w1, w2 };
    const float* lb[3] = { b0, b1, b2 };
    const int    lk[3] = { FIN, HID, HID };
    for (int L = 0; L < 3; L++) {
        int tiles = ((N + 15) / 16) * (HID / 16);
        k_gemm_f32_wmma<<<blks(tiles, 8), T, 0, stream>>>(lx[L], lw[L], hbuf, N, lk[L], HID);
        k_gcn_self<<<blks((long)N * HID, T), T, 0, stream>>>(hbuf, dinv, ly[L], N);
        k_gcn_edges<<<blks((long)E * 32, T), T, 0, stream>>>(ei, hbuf, dinv, ly[L], E);
        k_relu_bias<<<blks((long)N * HID, T), T, 0, stream>>>(ly[L], lb[L], N * HID);
    }
    // after layer 2, features are in xa

    // GAT: h = xa @ gat_w  (N x 256)
    {
        int tiles = ((N + 15) / 16) * ((HEADS * HID) / 16);
        k_gemm_f32_wmma<<<blks(tiles, 8), T, 0, stream>>>(xa, gat_w, hbuf, N, HID, HEADS * HID);
    }
    k_gat_scores<<<blks((long)N * HEADS, T), T, 0, stream>>>(hbuf, att_s, att_d, asrc, adst, N);
    k_uset<<<blks((long)N * HEADS, T), T, 0, stream>>>(menc, ENC_NEG_INF, N * HEADS);
    k_fset<<<blks((long)N * HEADS, T), T, 0, stream>>>(denom, 0.0f, N * HEADS);
    k_gat_max<<<blks((long)(E + N) * HEADS, T), T, 0, stream>>>(ei, asrc, adst, menc, E, N);
    k_gat_denom<<<blks((long)(E + N) * HEADS, T), T, 0, stream>>>(ei, asrc, adst, menc, denom, E, N);
    k_fset<<<blks((long)N * HID, T), T, 0, stream>>>(xb, 0.0f, N * HID);
    k_gat_aggr<<<blks((long)(E + N) * 32, T), T, 0, stream>>>(ei, hbuf, asrc, adst, menc, denom, xb, E, N);
    k_relu_bias<<<blks((long)N * HID, T), T, 0, stream>>>(xb, gat_b, N * HID);

    // pooling
    k_fset<<<blks((long)NGRAPH * HID, T), T, 0, stream>>>(pmean, 0.0f, NGRAPH * HID);
    k_uset<<<blks((long)NGRAPH * HID, T), T, 0, stream>>>(pmax, ENC_NEG_INF, NGRAPH * HID);
    k_fset<<<blks(NGRAPH, T), T, 0, stream>>>(pcnt, 0.0f, NGRAPH);
    k_pool_cnt<<<blks(N, T), T, 0, stream>>>(batch, pcnt, N);
    k_pool<<<blks((long)N * HID, T), T, 0, stream>>>(batch, xb, pmean, pmax, N);
    k_pool_finish<<<blks((long)NGRAPH * HID, T), T, 0, stream>>>(pmean, pmax, pcnt, gbuf, NGRAPH);

    // MLP head -> d_out
    k_mlp<<<blks(NGRAPH, T), T, 0, stream>>>(gbuf, l1w, l1b, l2w, l2b, out, NGRAPH);
}